// DocPolarBERTSelfAttention_27410481283221
// MI455X (gfx1250) — compile-verified
//
#include <hip/hip_runtime.h>
#include <hip/hip_bf16.h>

typedef __attribute__((ext_vector_type(2))) float v2f;
typedef __attribute__((ext_vector_type(8))) float v8f;

#define BATCH 2
#define S_DIM 512
#define D_DIM 768
#define NH    12
#define HD    64
#define SB    2          // query positions per block in attention kernel

// ---------------------------------------------------------------------------
// Kernel 1: QKV projection.  C[m,n] = sum_d hs[m,d] * W[n,d]  (+bias, *scale)
// One wave per 16x16 output tile, f32 WMMA 16x16x4 chained over K=768.
// Q -> (B,NH,S,HD) pre-scaled by 1/8;  K -> (B,NH,HD,S) TRANSPOSED;
// V -> (B,NH,S,HD).
// ---------------------------------------------------------------------------
__global__ __launch_bounds__(256) void qkv_proj_kernel(
    const float* __restrict__ hs,
    const float* __restrict__ Wq, const float* __restrict__ bq,
    const float* __restrict__ Wk, const float* __restrict__ bk,
    const float* __restrict__ Wv, const float* __restrict__ bv,
    float* __restrict__ qo, float* __restrict__ ko, float* __restrict__ vo)
{
    const int lane = threadIdx.x & 31;
    const int wave = threadIdx.x >> 5;
    const int NT = D_DIM / 16;               // 48 col tiles
    const int MT = (BATCH * S_DIM) / 16;     // 64 row tiles
    int gw  = blockIdx.x * 8 + wave;         // 0 .. 3*MT*NT-1 (exact)
    int p   = gw / (MT * NT);
    int rem = gw % (MT * NT);
    int mt  = rem / NT;
    int nt  = rem % NT;

    const float* W; const float* bias; float scale;
    if (p == 0)      { W = Wq; bias = bq; scale = 0.125f; } // 1/sqrt(64)
    else if (p == 1) { W = Wk; bias = bk; scale = 1.0f;   }
    else             { W = Wv; bias = bv; scale = 1.0f;   }

    const int m0   = mt * 16;
    const int n0   = nt * 16;
    const int half = lane >> 4;   // K half (A/B) / M half (D)
    const int l16  = lane & 15;

    // A: 16x4 of hs, lane holds (M=l16, K=2*half+v); B: (K=2*half+v, N=l16)
    const float* arow = hs + (size_t)(m0 + l16) * D_DIM + 2 * half;
    const float* brow = W  + (size_t)(n0 + l16) * D_DIM + 2 * half;

    v8f c = {};
    #pragma unroll 4
    for (int d0 = 0; d0 < D_DIM; d0 += 4) {
        v2f a = *(const v2f*)(arow + d0);
        v2f b = *(const v2f*)(brow + d0);
        c = __builtin_amdgcn_wmma_f32_16x16x4_f32(
                false, a, false, b, (short)0, c, false, false);
    }

    // D layout: element (M = r + 8*half, N = l16)
    #pragma unroll
    for (int r = 0; r < 8; ++r) {
        int M   = m0 + r + 8 * half;      // flat b*S + s
        int col = n0 + l16;               // flat n*HD + h
        int bb  = M / S_DIM, s = M % S_DIM;
        int n   = col / HD,  h = col % HD;
        float val = (c[r] + bias[col]) * scale;
        if (p == 0)
            qo[(((size_t)bb * NH + n) * S_DIM + s) * HD + h] = val;
        else if (p == 1)   // transposed: (B,NH,HD,S)
            ko[(((size_t)bb * NH + n) * HD + h) * S_DIM + s] = val;
        else
            vo[(((size_t)bb * NH + n) * S_DIM + s) * HD + h] = val;
    }
}

// ---------------------------------------------------------------------------
// Kernel 2: one workgroup per (b, pair-of-s).  Streams rel tensors ONCE
// (shared by all 12 heads via WMMA); K/V read once per block, reused by both
// query positions (2 FMAs per load).
// ---------------------------------------------------------------------------
__global__ __launch_bounds__(256) void attn_kernel(
    const float* __restrict__ qws,   // (B,NH,S,HD) pre-scaled
    const float* __restrict__ kws,   // (B,NH,HD,S) transposed
    const float* __restrict__ vws,   // (B,NH,S,HD)
    const float* __restrict__ rel0,  // (B,S,S,HD)
    const float* __restrict__ rel1,
    const float* __restrict__ rel2,
    const float* __restrict__ mask,  // (B,1,1,S)
    float* __restrict__ outp)        // (B,S,NH*HD)
{
    __shared__ float qs[SB][16][68];     // heads (padded to 16) x HD
    __shared__ float sc[SB][16][512];    // scores, rows 0..11 used

    const int tid  = threadIdx.x;
    const int lane = tid & 31;
    const int wave = tid >> 5;
    const int b    = blockIdx.x / (S_DIM / SB);
    const int s0   = (blockIdx.x % (S_DIM / SB)) * SB;

    // ---- Load QS[si][n][h]; zero-pad heads 12..15 -------------------------
    for (int i = tid; i < SB * 16 * HD; i += 256) {
        int si = i / (16 * HD);
        int r  = i % (16 * HD);
        int n = r / HD, h = r % HD;
        float v = 0.0f;
        if (n < NH) v = qws[(((size_t)b * NH + n) * S_DIM + (s0 + si)) * HD + h];
        qs[si][n][h] = v;
    }
    __syncthreads();

    const int half = lane >> 4;
    const int l16  = lane & 15;

    // ---- Rel-GEMM per query position (WMMA) -------------------------------
    for (int si = 0; si < SB; ++si) {
        // Preload 16 A-fragments (K chunks of 4 over HD=64)
        v2f afrag[16];
        #pragma unroll
        for (int st = 0; st < 16; ++st) {
            int h0 = 4 * st + 2 * half;
            afrag[st][0] = qs[si][l16][h0];
            afrag[st][1] = qs[si][l16][h0 + 1];
        }
        const size_t base_bs = ((size_t)b * S_DIM + (s0 + si)) * S_DIM;
        for (int kt = wave; kt < 32; kt += 8) {   // uniform per wave, EXEC all-1
            int k0 = kt * 16;
            size_t rrow = (base_bs + k0 + l16) * HD + 2 * half;
            const float* p0 = rel0 + rrow;
            const float* p1 = rel1 + rrow;
            const float* p2 = rel2 + rrow;
            v8f c = {};
            #pragma unroll
            for (int st = 0; st < 16; ++st) {
                int h0 = 4 * st;
                v2f bf = *(const v2f*)(p0 + h0);
                v2f t1 = *(const v2f*)(p1 + h0);
                v2f t2 = *(const v2f*)(p2 + h0);
                bf[0] += t1[0] + t2[0];
                bf[1] += t1[1] + t2[1];
                c = __builtin_amdgcn_wmma_f32_16x16x4_f32(
                        false, afrag[st], false, bf, (short)0, c, false, false);
            }
            #pragma unroll
            for (int r = 0; r < 8; ++r)
                sc[si][r + 8 * half][k0 + l16] = c[r];
        }
    }
    __syncthreads();

    // ---- QK^T: coalesced over kk (K transposed), both s per load + mask ---
    for (int idx = tid; idx < NH * S_DIM; idx += 256) {
        int n  = idx >> 9;        // wave-uniform
        int kk = idx & 511;       // consecutive across lanes -> coalesced
        const float* kT = kws + (((size_t)b * NH + n) * HD) * S_DIM + kk;
        float acc0 = 0.0f, acc1 = 0.0f;
        #pragma unroll 8
        for (int h = 0; h < HD; ++h) {
            float kv = kT[(size_t)h * S_DIM];
            acc0 += qs[0][n][h] * kv;      // wave-uniform LDS broadcast
            acc1 += qs[1][n][h] * kv;
        }
        float mval = mask[b * S_DIM + kk];
        sc[0][n][kk] += acc0 + mval;
        sc[1][n][kk] += acc1 + mval;
    }
    __syncthreads();

    // ---- Softmax (PB-Relax == standard stable softmax), one wave per row --
    for (int r = wave; r < SB * NH; r += 8) {
        int si = r / NH, n = r % NH;
        float m = -3.402823e38f;
        for (int kk = lane; kk < S_DIM; kk += 32) m = fmaxf(m, sc[si][n][kk]);
        #pragma unroll
        for (int off = 16; off > 0; off >>= 1) m = fmaxf(m, __shfl_xor(m, off, 32));
        float sum = 0.0f;
        for (int kk = lane; kk < S_DIM; kk += 32) {
            float e = __expf(sc[si][n][kk] - m);
            sc[si][n][kk] = e;
            sum += e;
        }
        #pragma unroll
        for (int off = 16; off > 0; off >>= 1) sum += __shfl_xor(sum, off, 32);
        float inv = 1.0f / sum;
        for (int kk = lane; kk < S_DIM; kk += 32) sc[si][n][kk] *= inv;
    }
    __syncthreads();

    // ---- ctx: coalesced over h; V loaded once, used for both s ------------
    for (int col = tid; col < NH * HD; col += 256) {
        int n = col / HD, h = col % HD;   // consecutive h across lanes
        const float* vp = vws + (((size_t)b * NH + n) * S_DIM) * HD + h;
        float acc0 = 0.0f, acc1 = 0.0f;
        for (int kk = 0; kk < S_DIM; ++kk) {
            float vv = vp[(size_t)kk * HD];
            acc0 += sc[0][n][kk] * vv;     // wave-uniform LDS broadcast
            acc1 += sc[1][n][kk] * vv;
        }
        outp[((size_t)b * S_DIM + s0)     * (NH * HD) + col] = acc0;
        outp[((size_t)b * S_DIM + s0 + 1) * (NH * HD) + col] = acc1;
    }
}

// ---------------------------------------------------------------------------
extern "C" void kernel_launch(void* const* d_in, const int* in_sizes, int n_in,
                              void* d_out, int out_size, void* d_ws, size_t ws_size,
                              hipStream_t stream) {
    const float* hs   = (const float*)d_in[0];
    const float* mask = (const float*)d_in[1];
    const float* rel0 = (const float*)d_in[2];
    const float* rel1 = (const float*)d_in[3];
    const float* rel2 = (const float*)d_in[4];
    const float* Wq   = (const float*)d_in[5];
    const float* bq   = (const float*)d_in[6];
    const float* Wk   = (const float*)d_in[7];
    const float* bk   = (const float*)d_in[8];
    const float* Wv   = (const float*)d_in[9];
    const float* bv   = (const float*)d_in[10];

    const size_t per = (size_t)BATCH * NH * S_DIM * HD;  // 786432 floats
    float* qws = (float*)d_ws;
    float* kws = qws + per;
    float* vws = kws + per;

    // 3 proj * 64 * 48 tiles = 9216 waves = 1152 blocks of 8 waves
    qkv_proj_kernel<<<1152, 256, 0, stream>>>(hs, Wq, bq, Wk, bk, Wv, bv,
                                              qws, kws, vws);
    attn_kernel<<<BATCH * (S_DIM / SB), 256, 0, stream>>>(qws, kws, vws,
                                                          rel0, rel1, rel2, mask,
                                                          (float*)d_out);
}